// ReconstructCoords_48412871361150
// MI455X (gfx1250) — compile-verified
//
#include <hip/hip_runtime.h>
#include <hip/hip_bf16.h>
#include <math.h>

// Problem constants (from reference)
#define NMAT   21
#define NSQ    441          // 21*21
#define KR     3
#define NSTEPS 10
#define WPB    8            // waves (=batches) per 256-thread block
#define TPB    256

typedef __attribute__((ext_vector_type(2))) float v2f;
typedef __attribute__((ext_vector_type(8))) float v8f;

__device__ __forceinline__ float hash_to_unit(unsigned x) {
  x ^= x >> 16; x *= 0x7feb352dU;
  x ^= x >> 15; x *= 0x846ca68bU;
  x ^= x >> 16;
  return ((float)(x >> 8)) * (2.0f / 16777216.0f) - 1.0f;   // (-1, 1)
}

__device__ __forceinline__ float wave_sum32(float v) {
  #pragma unroll
  for (int m = 16; m > 0; m >>= 1) v += __shfl_xor(v, m, 32);
  return v;                                                  // all lanes hold sum
}

// ---------------------------------------------------------------------------
// Kernel 1: Gram construction  A = mask * -0.5 * (D^2 - col/21 - row/21 + tot/441)
// one wave32 per batch, Dsq staged in LDS
// ---------------------------------------------------------------------------
__global__ void gram_kernel(const float* __restrict__ D,
                            const float* __restrict__ maskD,
                            float* __restrict__ A) {
  __shared__ float ds[WPB][NSQ];
  __shared__ float rsum[WPB][NMAT];   // sum over axis=2 (row sums)
  __shared__ float csum[WPB][NMAT];   // sum over axis=1 (col sums)
  const int lane = threadIdx.x & 31;
  const int w    = threadIdx.x >> 5;
  const size_t b = (size_t)blockIdx.x * WPB + w;
  const float* Db = D + b * NSQ;
  for (int idx = lane; idx < NSQ; idx += 32) {
    float d = Db[idx];
    ds[w][idx] = d * d;
  }
  __syncthreads();
  float rs = 0.0f;
  if (lane < NMAT) {
    float cs = 0.0f;
    #pragma unroll 1
    for (int j = 0; j < NMAT; ++j) {
      rs += ds[w][lane * NMAT + j];
      cs += ds[w][j * NMAT + lane];
    }
    rsum[w][lane] = rs;
    csum[w][lane] = cs;
  }
  const float total = wave_sum32((lane < NMAT) ? rs : 0.0f);
  __syncthreads();
  const float* Mb = maskD + b * NSQ;
  float* Ab = A + b * NSQ;
  const float inv_n  = 1.0f / 21.0f;
  const float inv_n2 = 1.0f / 441.0f;
  for (int idx = lane; idx < NSQ; idx += 32) {
    int i = idx / NMAT;
    int j = idx - i * NMAT;
    float v = ds[w][idx] - (csum[w][j] + rsum[w][i]) * inv_n + total * inv_n2;
    Ab[idx] = Mb[idx] * (-0.5f) * v;
  }
}

// ---------------------------------------------------------------------------
// Kernel 2: deterministic pseudo-random init of u + per-block partial sums of u^2
// ---------------------------------------------------------------------------
__global__ void init_kernel(float* __restrict__ u0,
                            float* __restrict__ partials, int rank) {
  __shared__ float ysq[WPB][NMAT];
  const int lane = threadIdx.x & 31;
  const int w    = threadIdx.x >> 5;
  const size_t b = (size_t)blockIdx.x * WPB + w;
  if (lane < NMAT) {
    unsigned seed = (unsigned)(b * 21u + (unsigned)lane) * 2654435761u
                  ^ ((unsigned)(rank + 1) * 0x9E3779B9u);
    float v = hash_to_unit(seed);
    u0[b * NMAT + lane] = v;
    ysq[w][lane] = v * v;
  }
  __syncthreads();
  if (w == 0 && lane < NMAT) {
    float acc = 0.0f;
    #pragma unroll
    for (int ww = 0; ww < WPB; ++ww) acc += ysq[ww][lane];
    partials[(size_t)blockIdx.x * NMAT + lane] = acc;
  }
}

// ---------------------------------------------------------------------------
// Kernel 3: deterministic tree reduce of per-block partials -> sums[21]
// block n handles position n with a fixed-order strided sum + tree
// ---------------------------------------------------------------------------
__global__ void reduce_kernel(const float* __restrict__ partials,
                              float* __restrict__ sums, int nblk) {
  __shared__ float buf[TPB];
  const int n = blockIdx.x;            // 0..20
  float acc = 0.0f;
  for (int k = threadIdx.x; k < nblk; k += TPB)
    acc += partials[(size_t)k * NMAT + n];
  buf[threadIdx.x] = acc;
  __syncthreads();
  #pragma unroll
  for (int s = TPB / 2; s > 0; s >>= 1) {
    if ((int)threadIdx.x < s) buf[threadIdx.x] += buf[threadIdx.x + s];
    __syncthreads();
  }
  if (threadIdx.x == 0) sums[n] = buf[0];
}

// ---------------------------------------------------------------------------
// Kernel 4: one power-iteration step, one wave32 per batch.
//   u_n = u_in / max(sqrt(sums), 1e-3)            (batch-axis normalize)
//   y   = A @ u_n - sum_j (u_j . u_n) u_j          (deflated matvec, WMMA f32)
//   u_out = y ; partials[block][n] = sum_waves y[n]^2
// WMMA mapping: 21x21 padded to 2 row-tiles (M 0..15 / 16..31, rows>=21 zero)
// x 6 K-chunks of 4 (K 0..23, cols>=21 zero) using V_WMMA_F32_16X16X4_F32.
// B operand = u replicated across all 16 N columns, so D[m][n] == y[m] for all n.
// ---------------------------------------------------------------------------
__global__ void power_step_kernel(const float* __restrict__ A,
                                  const float* __restrict__ u_in,
                                  float* __restrict__ u_out,
                                  const float* __restrict__ sums,
                                  float* __restrict__ partials,
                                  const float* __restrict__ X,
                                  int rank) {
  __shared__ float uarea[WPB][24];
  __shared__ float yarea[WPB][24];
  __shared__ float ysq[WPB][24];
  const int lane = threadIdx.x & 31;
  const int w    = threadIdx.x >> 5;
  const size_t b = (size_t)blockIdx.x * WPB + w;

  // normalize over batch axis
  float un = 0.0f;
  if (lane < NMAT) {
    float nrm = fmaxf(sqrtf(sums[lane]), 1e-3f);
    un = u_in[b * NMAT + lane] / nrm;
  }
  if (lane < 24) uarea[w][lane] = (lane < NMAT) ? un : 0.0f;
  __syncthreads();

  // deflation dots (rank is wave-uniform)
  float xj0 = 0.0f, xj1 = 0.0f, d0 = 0.0f, d1 = 0.0f;
  if (rank > 0) {
    if (lane < NMAT) xj0 = X[b * (NMAT * KR) + lane * KR + 0];
    d0 = wave_sum32(xj0 * un);
  }
  if (rank > 1) {
    if (lane < NMAT) xj1 = X[b * (NMAT * KR) + lane * KR + 1];
    d1 = wave_sum32(xj1 * un);
  }

  // --- WMMA matvec: EXEC is all-ones here; OOB handled by clamp+zero-mult ---
  const float* Ab = A + b * NSQ;
  const int   mrow = lane & 15;                 // A layout: M = lane&15
  const int   koff = (lane >> 4) << 1;          // lanes 16-31 carry K+2
  const int   mb   = 16 + mrow;
  const float mbv  = (mb < NMAT) ? 1.0f : 0.0f;
  const int   mbc  = (mb < NMAT) ? mb : (NMAT - 1);
  v8f ct = {};
  v8f cb = {};
  #pragma unroll
  for (int c = 0; c < 6; ++c) {
    const int   k0  = 4 * c + koff;
    const int   k1  = k0 + 1;
    const float p0  = (k0 < NMAT) ? 1.0f : 0.0f;
    const float p1  = (k1 < NMAT) ? 1.0f : 0.0f;
    const int   k0c = (k0 < NMAT) ? k0 : (NMAT - 1);
    const int   k1c = (k1 < NMAT) ? k1 : (NMAT - 1);
    v2f bv; bv.x = uarea[w][k0];  bv.y = uarea[w][k1];      // zero-padded >= 21
    v2f at; at.x = Ab[mrow * NMAT + k0c] * p0;
            at.y = Ab[mrow * NMAT + k1c] * p1;
    ct = __builtin_amdgcn_wmma_f32_16x16x4_f32(false, at, false, bv,
                                               (short)0, ct, false, false);
    v2f ab; ab.x = Ab[mbc * NMAT + k0c] * (p0 * mbv);
            ab.y = Ab[mbc * NMAT + k1c] * (p1 * mbv);
    cb = __builtin_amdgcn_wmma_f32_16x16x4_f32(false, ab, false, bv,
                                               (short)0, cb, false, false);
  }

  // gather y: C/D layout -> lane0 vgpr j = D[j][0]; lane16 vgpr j = D[8+j][0]
  if (lane == 0) {
    #pragma unroll
    for (int j = 0; j < 8; ++j) yarea[w][j] = ct[j];
    #pragma unroll
    for (int j = 0; j < 5; ++j) yarea[w][16 + j] = cb[j];
  }
  if (lane == 16) {
    #pragma unroll
    for (int j = 0; j < 8; ++j) yarea[w][8 + j] = ct[j];
  }
  __syncthreads();

  float yv = 0.0f;
  if (lane < NMAT) {
    yv = yarea[w][lane] - d0 * xj0 - d1 * xj1;
    u_out[b * NMAT + lane] = yv;
    ysq[w][lane] = yv * yv;
  }
  __syncthreads();
  if (w == 0 && lane < NMAT) {
    float acc = 0.0f;
    #pragma unroll
    for (int ww = 0; ww < WPB; ++ww) acc += ysq[ww][lane];
    partials[(size_t)blockIdx.x * NMAT + lane] = acc;
  }
}

// ---------------------------------------------------------------------------
// Kernel 5: eig scaling  X[b,:,rank] = u / (sum_n u^2 + 0.01)^0.25
// ---------------------------------------------------------------------------
__global__ void finalize_kernel(const float* __restrict__ u_in,
                                float* __restrict__ X, int rank) {
  const int lane = threadIdx.x & 31;
  const int w    = threadIdx.x >> 5;
  const size_t b = (size_t)blockIdx.x * WPB + w;
  float uv = (lane < NMAT) ? u_in[b * NMAT + lane] : 0.0f;
  float eig = wave_sum32(uv * uv);
  float sc = rsqrtf(sqrtf(eig + 0.01f));   // (eig+eps2)^-0.25
  if (lane < NMAT) X[b * (NMAT * KR) + lane * KR + rank] = uv * sc;
}

// ---------------------------------------------------------------------------
// Host-side orchestration (graph-capture safe: launches only)
// ---------------------------------------------------------------------------
extern "C" void kernel_launch(void* const* d_in, const int* in_sizes, int n_in,
                              void* d_out, int out_size, void* d_ws, size_t ws_size,
                              hipStream_t stream) {
  (void)n_in; (void)out_size; (void)ws_size;
  const float* D     = (const float*)d_in[0];
  // d_in[1] = W (unused by reference), d_in[2] = batch_size scalar (unused)
  const float* maskD = (const float*)d_in[3];
  float* X = (float*)d_out;                       // [B, 21, 3]

  const int B    = in_sizes[0] / NSQ;             // 65536
  const int nblk = B / WPB;                       // 8192

  // workspace carve (floats): A | u_a | u_b | partials | sums
  float* A        = (float*)d_ws;                              // B*441
  float* ua       = A  + (size_t)B * NSQ;                      // B*21
  float* ub       = ua + (size_t)B * NMAT;                     // B*21
  float* partials = ub + (size_t)B * NMAT;                     // nblk*21
  float* sums     = partials + (size_t)nblk * NMAT;            // 21 (+pad)

  gram_kernel<<<nblk, TPB, 0, stream>>>(D, maskD, A);

  for (int rank = 0; rank < KR; ++rank) {
    init_kernel<<<nblk, TPB, 0, stream>>>(ua, partials, rank);
    reduce_kernel<<<NMAT, TPB, 0, stream>>>(partials, sums, nblk);
    float* ucur = ua;
    float* unext = ub;
    for (int step = 0; step < NSTEPS; ++step) {
      power_step_kernel<<<nblk, TPB, 0, stream>>>(A, ucur, unext, sums,
                                                  partials, X, rank);
      if (step + 1 < NSTEPS)
        reduce_kernel<<<NMAT, TPB, 0, stream>>>(partials, sums, nblk);
      float* t = ucur; ucur = unext; unext = t;
    }
    finalize_kernel<<<nblk, TPB, 0, stream>>>(ucur, X, rank);
  }
}